// AggregateLayer_61168924229860
// MI455X (gfx1250) — compile-verified
//
#include <hip/hip_runtime.h>
#include <math.h>

// AggregateLayer on MI455X (gfx1250, wave32).
// B=8192, N=64, D=128, all fp32.
// Memory bound: 256MB of `pref` streamed once -> ~11us floor at 23.3 TB/s.
// Strategy: 1 block (4 waves) per batch; stage pref[b] (32KB) into LDS via the
// CDNA5 async global->LDS pipe (GLOBAL_LOAD_ASYNC_TO_LDS_B128, ASYNCcnt) when
// the toolchain exposes it; compute the 64x128 @ 128 matvec with
// V_WMMA_F32_16X16X4_F32 (full fp32 matrix path); softmax + the 64-term
// weighted sum in VALU (negligible).

typedef __attribute__((ext_vector_type(2))) float v2f;
typedef __attribute__((ext_vector_type(8))) float v8f;
typedef __attribute__((ext_vector_type(4))) int   v4i;

#define NN 64
#define DD 128
// Row stride padding 128 -> 132 floats: keeps every LDS row 16B-aligned
// (132*4 = 528 = 33*16) so 16B LDS stores are legal, and puts consecutive
// rows 4 banks apart -> conflict-free A-tile fetch (lanes 0..15 read 16
// different rows at the same column) and conflict-free phase-4 column reads.
#define LDS_STRIDE 132

#if __has_builtin(__builtin_amdgcn_global_load_async_to_lds_b128) && \
    __has_builtin(__builtin_amdgcn_s_wait_asynccnt)
#define USE_ASYNC_LDS 1
#else
#define USE_ASYNC_LDS 0
#endif

__global__ __launch_bounds__(128) void agg_softmax_wmma_kernel(
    const float* __restrict__ pref,    // (B, N, D)
    const float* __restrict__ cvec,    // (B, 1, D)
    const float* __restrict__ t_pref,  // (B, 1, N)
    const float* __restrict__ t_c,     // (B, 1)
    float* __restrict__ out)           // (B, 1, D)
{
    __shared__ float sP[NN * LDS_STRIDE];  // 33,792 B
    __shared__ float sC[DD];
    __shared__ float sDot[NN];
    __shared__ float sE[NN];

    const int b    = blockIdx.x;
    const int tid  = threadIdx.x;      // 0..127
    const int lane = tid & 31;
    const int wave = tid >> 5;         // 0..3

    // ---------- stage pref[b] (64x128 f32 = 32KB) into LDS ----------
    const float4* gp = (const float4*)(pref + (size_t)b * NN * DD);
    #pragma unroll
    for (int i = 0; i < (NN * DD / 4) / 128; ++i) {   // 16 x 16B per thread
        int g   = tid + i * 128;       // float4 index within the tile
        int row = g >> 5;              // 32 float4 per row
        int c4  = g & 31;
#if USE_ASYNC_LDS
        // cache/memory -> LDS directly, no VGPR round-trip; ASYNCcnt-tracked
        __builtin_amdgcn_global_load_async_to_lds_b128(
            (__attribute__((address_space(1))) v4i*)(gp + g),
            (__attribute__((address_space(3))) v4i*)&sP[row * LDS_STRIDE + c4 * 4],
            /*offset=*/0, /*cpol=*/0);
#else
        float4 v = gp[g];                              // global_load_b128
        *(float4*)&sP[row * LDS_STRIDE + c4 * 4] = v;  // ds_store_b128
#endif
    }
    sC[tid] = cvec[(size_t)b * DD + tid];
#if USE_ASYNC_LDS
    __builtin_amdgcn_s_wait_asynccnt(0);   // this wave's async LDS writes done
#endif
    __syncthreads();                       // publish LDS across the 4 waves

    // ---------- phase 2: dot_w = pref(64x128) x c(128) via WMMA f32 16x16x4
    // Wave w owns rows 16w..16w+15 (M). B-tile columns are all replicated
    // with the c[k..k+3] slice, so D is column-replicated and only the
    // K <-> (vgpr, lane-half) mapping matters.
    // ISA A layout (16x4 f32): VGPR0 = K{0 | 2}, VGPR1 = K{1 | 3}, M = lane&15.
    const int   mRow  = wave * 16 + (lane & 15);
    const int   kHalf = (lane >> 4) << 1;              // 0 or 2
    const float* rowp = &sP[mRow * LDS_STRIDE];
    v8f acc = {};                                      // C starts at 0
    #pragma unroll 4
    for (int k = 0; k < DD; k += 4) {
        v2f a, bb;
        a.x  = rowp[k + kHalf];
        a.y  = rowp[k + kHalf + 1];
        bb.x = sC[k + kHalf];
        bb.y = sC[k + kHalf + 1];
        // (neg_a, A, neg_b, B, c_mod, C, reuse_a, reuse_b)
        acc = __builtin_amdgcn_wmma_f32_16x16x4_f32(
            false, a, false, bb, (short)0, acc, false, false);
    }
    // D layout: acc[v], lanes 0-15 -> M=v ; lanes 16-31 -> M=v+8 (N replicated)
    if ((lane & 15) == 0) {
        const int moff = (lane >> 4) * 8;
        #pragma unroll
        for (int v = 0; v < 8; ++v)
            sDot[wave * 16 + moff + v] = acc[v];
    }
    __syncthreads();

    // ---------- phase 3: score = dot + 1/|t_pref - t_c|, softmax over N=64
    if (tid < NN) {
        float tw = 1.0f / fabsf(t_pref[(size_t)b * NN + tid] - t_c[b]);
        sDot[tid] += tw;
    }
    __syncthreads();
    float mx = -3.402823466e38f;
    #pragma unroll 8
    for (int j = 0; j < NN; ++j) mx = fmaxf(mx, sDot[j]);  // broadcast reads
    if (tid < NN) sE[tid] = __expf(sDot[tid] - mx);
    __syncthreads();
    float sum = 0.0f;
    #pragma unroll 8
    for (int j = 0; j < NN; ++j) sum += sE[j];
    const float inv = 1.0f / sum;

    // ---------- phase 4: u[d] = inv * sum_j e[j] * pref[j][d] (VALU, cheap)
    float u = 0.0f;
    #pragma unroll 8
    for (int j = 0; j < NN; ++j)
        u = fmaf(sE[j], sP[j * LDS_STRIDE + tid], u);
    out[(size_t)b * DD + tid] = u * inv;
}

extern "C" void kernel_launch(void* const* d_in, const int* in_sizes, int n_in,
                              void* d_out, int out_size, void* d_ws, size_t ws_size,
                              hipStream_t stream) {
    (void)n_in; (void)out_size; (void)d_ws; (void)ws_size;
    const float* pref   = (const float*)d_in[0];
    const float* cvec   = (const float*)d_in[1];
    const float* t_pref = (const float*)d_in[2];
    const float* t_c    = (const float*)d_in[3];
    float* out = (float*)d_out;
    const int B = in_sizes[3];   // t_c has B elements
    agg_softmax_wmma_kernel<<<dim3(B), dim3(128), 0, stream>>>(
        pref, cvec, t_pref, t_c, out);
}